// Generate_G_from_H_16011638079612
// MI455X (gfx1250) — compile-verified
//
#include <hip/hip_runtime.h>

typedef __attribute__((ext_vector_type(2))) float v2f;
typedef __attribute__((ext_vector_type(8))) float v8f;

#define NN 8192
#define KSTEP 32
#define LDSS 34                 // stage row stride (32+2): 8B-aligned v2f frags, conflict-free
#define STAGEF (128 * LDSS)     // 4352 floats per stage matrix
#define TS 132                  // transpose row stride (128+4): 16B-aligned float4 rows
#define SMEMF (4 * STAGEF)      // 17408 floats = 69.6KB (>= 128*TS = 16896 for sT alias)

// ---------------- degree reductions (deterministic, no atomics) ----------------

__global__ __launch_bounds__(256) void rowsum_kernel(const float* __restrict__ H,
                                                     float* __restrict__ DV) {
    __shared__ float red[256];
    const int row = blockIdx.x;
    const int t = threadIdx.x;
    const float4* Hrow = (const float4*)(H + (size_t)row * NN);
    float s = 0.0f;
    #pragma unroll
    for (int i = 0; i < 8; ++i) {
        float4 v = Hrow[t + i * 256];
        s += (v.x + v.y) + (v.z + v.w);
    }
    red[t] = s;
    __syncthreads();
    for (int off = 128; off > 0; off >>= 1) {
        if (t < off) red[t] += red[t + off];
        __syncthreads();
    }
    if (t == 0) DV[row] = red[0];
}

__global__ __launch_bounds__(256) void colsum_part_kernel(const float* __restrict__ H,
                                                          float* __restrict__ DEpart) {
    const int c = blockIdx.x * 256 + threadIdx.x;
    const int r0 = blockIdx.y * 512;
    float s = 0.0f;
    for (int r = 0; r < 512; ++r)
        s += H[(size_t)(r0 + r) * NN + c];
    DEpart[(size_t)blockIdx.y * NN + c] = s;
}

__global__ __launch_bounds__(256) void colsum_final_kernel(const float* __restrict__ DEpart,
                                                           float* __restrict__ DE) {
    const int c = blockIdx.x * 256 + threadIdx.x;
    float s = 0.0f;
    #pragma unroll
    for (int p = 0; p < 16; ++p)
        s += DEpart[(size_t)p * NN + c];
    DE[c] = s;
}

__global__ __launch_bounds__(256) void inv_kernel(const float* __restrict__ DV,
                                                  const float* __restrict__ DE,
                                                  float* __restrict__ invDV,
                                                  float* __restrict__ invDE) {
    const int i = blockIdx.x * 256 + threadIdx.x;
    const float dv = DV[i];
    const float de = DE[i];
    invDV[i] = (dv > 0.0f) ? (1.0f / sqrtf(dv)) : 0.0f;
    invDE[i] = (de > 0.0f) ? (1.0f / de) : 0.0f;
}

// ---------------- fp32 WMMA symmetric GEMM, double-buffered ----------------
// G[i,j] = invDV[i]*invDV[j] * sum_e (H[i,e]*invDE[e]) * H[j,e]   (symmetric)
__global__ __launch_bounds__(256) void hgnn_gemm_kernel(const float* __restrict__ H,
                                                        const float* __restrict__ invDV,
                                                        const float* __restrict__ invDE,
                                                        float* __restrict__ G) {
    __shared__ float smem[SMEMF];
    float* const b0A = smem;
    float* const b0B = smem + STAGEF;
    float* const b1A = smem + 2 * STAGEF;
    float* const b1B = smem + 3 * STAGEF;
    float* const sT  = smem;            // epilogue alias (stages dead by then)

    // L2-locality swizzle: 8-wide column bands, column-fastest within band.
    const int id   = blockIdx.y * 64 + blockIdx.x;
    const int band = id >> 9;
    const int rem  = id & 511;
    const int bx   = band * 8 + (rem & 7);    // block col (j)
    const int by   = rem >> 3;                // block row (i)
    if (by > bx) return;                      // upper triangle only (uniform exit)

    const int t     = threadIdx.x;
    const int lane  = t & 31;
    const int wave  = t >> 5;
    const int waveM = wave >> 2;       // 0..1
    const int waveN = wave & 3;        // 0..3
    const int half  = lane >> 4;
    const int l16   = lane & 15;

    const int rowBase = by * 128;
    const int colBase = bx * 128;

    // Staging: q = t&7 is loop-invariant (256 % 8 == 0) -> one invDE vec per stage.
    const int q    = t & 7;
    const int row0 = t >> 3;           // 0..31, rows row0 + 32*i

    float4 rA[4], rB[4], rD;
    auto loadRegs = [&](int kk) {
        const int e = kk + q * 4;
        rD = *(const float4*)(invDE + e);
        #pragma unroll
        for (int i = 0; i < 4; ++i) {
            const int row = row0 + 32 * i;
            rA[i] = *(const float4*)(H + (size_t)(rowBase + row) * NN + e);
            rB[i] = *(const float4*)(H + (size_t)(colBase + row) * NN + e);
        }
    };
    auto storeStage = [&](float* dA, float* dB) {
        #pragma unroll
        for (int i = 0; i < 4; ++i) {
            const int row = row0 + 32 * i;
            float* pa = dA + row * LDSS + q * 4;
            pa[0] = rA[i].x * rD.x; pa[1] = rA[i].y * rD.y;
            pa[2] = rA[i].z * rD.z; pa[3] = rA[i].w * rD.w;
            float* pb = dB + row * LDSS + q * 4;
            pb[0] = rB[i].x; pb[1] = rB[i].y; pb[2] = rB[i].z; pb[3] = rB[i].w;
        }
    };

    v8f acc[4][2];
    const v8f vzero = {0.f, 0.f, 0.f, 0.f, 0.f, 0.f, 0.f, 0.f};
    #pragma unroll
    for (int tm = 0; tm < 4; ++tm)
        #pragma unroll
        for (int tn = 0; tn < 2; ++tn)
            acc[tm][tn] = vzero;

    auto computeStage = [&](const float* cA, const float* cB) {
        #pragma unroll
        for (int ks = 0; ks < KSTEP; ks += 4) {
            v2f a[4], b[2];
            #pragma unroll
            for (int tm = 0; tm < 4; ++tm)   // A frag: lane L, vgpr v -> A[L%16, 2*(L/16)+v]
                a[tm] = *(const v2f*)&cA[(waveM * 64 + tm * 16 + l16) * LDSS + ks + half * 2];
            #pragma unroll
            for (int tn = 0; tn < 2; ++tn)   // B staged transposed: same addressing
                b[tn] = *(const v2f*)&cB[(waveN * 32 + tn * 16 + l16) * LDSS + ks + half * 2];
            #pragma unroll
            for (int tm = 0; tm < 4; ++tm)
                #pragma unroll
                for (int tn = 0; tn < 2; ++tn)
                    acc[tm][tn] = __builtin_amdgcn_wmma_f32_16x16x4_f32(
                        false, a[tm], false, b[tn],
                        (short)0, acc[tm][tn], false, false);
        }
    };

    // Prologue: fill buffer 0.
    loadRegs(0);
    storeStage(b0A, b0B);
    __syncthreads();

    // Double-buffered main loop: one barrier per stage; global loads for the
    // next stage overlap the current stage's WMMA stream.
    for (int kk = 0; kk < NN; kk += 2 * KSTEP) {
        loadRegs(kk + KSTEP);            // always in range (NN % 64 == 0)
        computeStage(b0A, b0B);
        storeStage(b1A, b1B);            // prev readers of buf1 sync'd last iter
        __syncthreads();

        const bool more = (kk + 2 * KSTEP < NN);
        if (more) loadRegs(kk + 2 * KSTEP);
        computeStage(b1A, b1B);
        if (more) storeStage(b0A, b0B);
        __syncthreads();
    }

    // ---- Epilogue: scale by invDV[i]*invDV[j]; coalesced stores via LDS bounce ----
    float sj[2];
    #pragma unroll
    for (int tn = 0; tn < 2; ++tn)
        sj[tn] = invDV[colBase + waveN * 32 + tn * 16 + l16];
    #pragma unroll
    for (int tm = 0; tm < 4; ++tm) {
        const int giBase = rowBase + waveM * 64 + tm * 16 + half * 8;
        #pragma unroll
        for (int r = 0; r < 8; ++r) {
            const float si = invDV[giBase + r];
            #pragma unroll
            for (int tn = 0; tn < 2; ++tn)
                acc[tm][tn][r] *= si * sj[tn];
        }
    }

    // Pass 1: normal tile.
    #pragma unroll
    for (int tm = 0; tm < 4; ++tm)
        #pragma unroll
        for (int tn = 0; tn < 2; ++tn)
            #pragma unroll
            for (int r = 0; r < 8; ++r) {
                const int m = waveM * 64 + tm * 16 + half * 8 + r;
                const int n = waveN * 32 + tn * 16 + l16;
                sT[m * TS + n] = acc[tm][tn][r];
            }
    __syncthreads();
    #pragma unroll
    for (int i = 0; i < 16; ++i) {
        const int idx = t + i * 256;
        const int row = idx >> 5;
        const int qq  = idx & 31;
        const float4 v = *(const float4*)&sT[row * TS + qq * 4];
        *(float4*)(G + (size_t)(rowBase + row) * NN + colBase + qq * 4) = v;
    }

    // Pass 2: mirror tile G[j,i]; skipped on diagonal blocks (determinism).
    if (bx != by) {
        __syncthreads();
        #pragma unroll
        for (int tm = 0; tm < 4; ++tm)
            #pragma unroll
            for (int tn = 0; tn < 2; ++tn)
                #pragma unroll
                for (int r = 0; r < 8; ++r) {
                    const int m = waveM * 64 + tm * 16 + half * 8 + r;
                    const int n = waveN * 32 + tn * 16 + l16;
                    sT[n * TS + m] = acc[tm][tn][r];
                }
        __syncthreads();
        #pragma unroll
        for (int i = 0; i < 16; ++i) {
            const int idx = t + i * 256;
            const int row = idx >> 5;
            const int qq  = idx & 31;
            const float4 v = *(const float4*)&sT[row * TS + qq * 4];
            *(float4*)(G + (size_t)(colBase + row) * NN + rowBase + qq * 4) = v;
        }
    }
}

// ---------------- launcher ----------------
extern "C" void kernel_launch(void* const* d_in, const int* in_sizes, int n_in,
                              void* d_out, int out_size, void* d_ws, size_t ws_size,
                              hipStream_t stream) {
    (void)in_sizes; (void)n_in; (void)out_size; (void)ws_size;
    const float* H = (const float*)d_in[0];
    float* G = (float*)d_out;

    float* ws     = (float*)d_ws;
    float* DV     = ws;                 // 8192
    float* DE     = ws + NN;            // 8192
    float* invDV  = ws + 2 * NN;        // 8192
    float* invDE  = ws + 3 * NN;        // 8192
    float* DEpart = ws + 4 * NN;        // 16 * 8192

    rowsum_kernel<<<NN, 256, 0, stream>>>(H, DV);
    colsum_part_kernel<<<dim3(NN / 256, 16), 256, 0, stream>>>(H, DEpart);
    colsum_final_kernel<<<NN / 256, 256, 0, stream>>>(DEpart, DE);
    inv_kernel<<<NN / 256, 256, 0, stream>>>(DV, DE, invDV, invDE);
    hgnn_gemm_kernel<<<dim3(64, 64), 256, 0, stream>>>(H, invDV, invDE, G);
}